// GraphConvolutionalEncoder_70892730188260
// MI455X (gfx1250) — compile-verified
//
#include <hip/hip_runtime.h>
#include <hip/hip_bf16.h>
#include <cstdint>

typedef float v2f __attribute__((ext_vector_type(2)));
typedef float v8f __attribute__((ext_vector_type(8)));

// ---------------------------------------------------------------------------
// Dense transform: out[N x Dout] = (RELU ? relu(A) : A)[N x K] @ W[K x Dout]
// One wave computes a 32(M) x 64(N) tile via V_WMMA_F32_16X16X4_F32:
// two 16-row subtiles share the 4 B fragments (8 WMMA per 10 VMEM loads).
// RELU is a template parameter so the hot loop has no branches; all
// addresses advance by pointer increment (no per-iteration 64-bit mul).
// ---------------------------------------------------------------------------
template <bool RELU>
__global__ __launch_bounds__(256) void gcn_gemm_wmma_f32(
    const float* __restrict__ A, const float* __restrict__ W,
    float* __restrict__ out, int N, int K, int Dout)
{
    const int lane         = threadIdx.x & 31;
    const int waveInBlock  = threadIdx.x >> 5;
    const int wavesPerBlk  = blockDim.x >> 5;
    const int waveId       = blockIdx.x * wavesPerBlk + waveInBlock;
    const int nTiles64     = Dout >> 6;            // 64-wide N tiles
    const int mt           = waveId / nTiles64;
    const int nt           = waveId - mt * nTiles64;
    const int m0           = mt * 32;
    if (m0 >= N) return;                           // wave-uniform exit

    const int n0   = nt * 64;
    const int half = lane >> 4;                    // 0: K pair {0,1}; 1: {2,3}
    const int lid  = lane & 15;

    // A row for each 16-row subtile; subtile 0 rows always valid (N % 16 == 0),
    // subtile 1 may be past the end on the last tile -> clamp load, skip store.
    const bool tile1Valid = (m0 + 16) < N;
    const int  rA0 = m0 + lid;
    const int  rA1 = tile1Valid ? (m0 + 16 + lid) : (N - 1);

    v8f acc00 = {}, acc01 = {}, acc02 = {}, acc03 = {};   // rows m0..m0+15
    v8f acc10 = {}, acc11 = {}, acc12 = {}, acc13 = {};   // rows m0+16..m0+31

    const float* arow0 = A + (size_t)rA0 * K + 2 * half;
    const float* arow1 = A + (size_t)rA1 * K + 2 * half;
    // B pointers: rows (k0 + 2*half) and (k0 + 2*half + 1), column n0 + lid.
    const float* wrow0 = W + (size_t)(2 * half) * Dout + n0 + lid;
    const float* wrow1 = wrow0 + Dout;
    const size_t wStep = (size_t)4 * Dout;         // advance 4 K-rows per step

    for (int k0 = 0; k0 < K; k0 += 4) {
        // A fragments: 16x4 f32 striped per ISA layout
        float2 av0 = *reinterpret_cast<const float2*>(arow0);
        float2 av1 = *reinterpret_cast<const float2*>(arow1);
        arow0 += 4;
        arow1 += 4;
        v2f a0, a1;
        if (RELU) {
            a0[0] = fmaxf(av0.x, 0.0f); a0[1] = fmaxf(av0.y, 0.0f);
            a1[0] = fmaxf(av1.x, 0.0f); a1[1] = fmaxf(av1.y, 0.0f);
        } else {
            a0[0] = av0.x; a0[1] = av0.y;
            a1[0] = av1.x; a1[1] = av1.y;
        }

        // B fragments: 4x16 f32, row K striped across lanes; shared by both subtiles
        v2f b0; b0[0] = wrow0[0];  b0[1] = wrow1[0];
        v2f b1; b1[0] = wrow0[16]; b1[1] = wrow1[16];
        v2f b2; b2[0] = wrow0[32]; b2[1] = wrow1[32];
        v2f b3; b3[0] = wrow0[48]; b3[1] = wrow1[48];
        wrow0 += wStep;
        wrow1 += wStep;

        acc00 = __builtin_amdgcn_wmma_f32_16x16x4_f32(false, a0, false, b0, (short)0, acc00, false, false);
        acc01 = __builtin_amdgcn_wmma_f32_16x16x4_f32(false, a0, false, b1, (short)0, acc01, false, false);
        acc02 = __builtin_amdgcn_wmma_f32_16x16x4_f32(false, a0, false, b2, (short)0, acc02, false, false);
        acc03 = __builtin_amdgcn_wmma_f32_16x16x4_f32(false, a0, false, b3, (short)0, acc03, false, false);
        acc10 = __builtin_amdgcn_wmma_f32_16x16x4_f32(false, a1, false, b0, (short)0, acc10, false, false);
        acc11 = __builtin_amdgcn_wmma_f32_16x16x4_f32(false, a1, false, b1, (short)0, acc11, false, false);
        acc12 = __builtin_amdgcn_wmma_f32_16x16x4_f32(false, a1, false, b2, (short)0, acc12, false, false);
        acc13 = __builtin_amdgcn_wmma_f32_16x16x4_f32(false, a1, false, b3, (short)0, acc13, false, false);
    }

    // C/D layout: VGPR r holds row (sub-base + r + 8*half), col n0 + lid (+16/acc)
    {
        float* orow = out + (size_t)(m0 + 8 * half) * Dout + n0 + lid;
#pragma unroll
        for (int r = 0; r < 8; ++r) {
            float* p = orow + (size_t)r * Dout;
            p[0]  = acc00[r];
            p[16] = acc01[r];
            p[32] = acc02[r];
            p[48] = acc03[r];
        }
    }
    if (tile1Valid) {
        float* orow = out + (size_t)(m0 + 16 + 8 * half) * Dout + n0 + lid;
#pragma unroll
        for (int r = 0; r < 8; ++r) {
            float* p = orow + (size_t)r * Dout;
            p[0]  = acc10[r];
            p[16] = acc11[r];
            p[32] = acc12[r];
            p[48] = acc13[r];
        }
    }
}

// ---------------------------------------------------------------------------
// agg[i, j] = bias[j]   (D is a power of two)
// ---------------------------------------------------------------------------
__global__ __launch_bounds__(256) void gcn_fill_bias(
    float* __restrict__ agg, const float* __restrict__ bias,
    long long total, int dMask)
{
    long long i = (long long)blockIdx.x * blockDim.x + threadIdx.x;
    if (i < total) agg[i] = bias[(int)(i & dMask)];
}

// ---------------------------------------------------------------------------
// Edge scatter: agg[dst[e], :] += support[src[e], :]
// One thread handles 4 consecutive features of one edge (float4 gather,
// 4x global_atomic_add_f32). Consecutive threads -> consecutive features.
// ---------------------------------------------------------------------------
__global__ __launch_bounds__(256) void gcn_scatter_add(
    const float* __restrict__ support, const int* __restrict__ src,
    const int* __restrict__ dst, float* __restrict__ agg,
    long long total, int fShift, int fMask, int D)
{
    long long tid = (long long)blockIdx.x * blockDim.x + threadIdx.x;
    if (tid >= total) return;
    const int e = (int)(tid >> fShift);
    const int f = ((int)tid & fMask) << 2;          // feature offset (x4)
    const int s = src[e];
    const int d = dst[e];
    const float4 v = *reinterpret_cast<const float4*>(support + (size_t)s * D + f);
    float* p = agg + (size_t)d * D + f;
    unsafeAtomicAdd(p + 0, v.x);
    unsafeAtomicAdd(p + 1, v.y);
    unsafeAtomicAdd(p + 2, v.z);
    unsafeAtomicAdd(p + 3, v.w);
}

// ---------------------------------------------------------------------------
// In-place ReLU
// ---------------------------------------------------------------------------
__global__ __launch_bounds__(256) void gcn_relu(float* __restrict__ x, long long total)
{
    long long i = (long long)blockIdx.x * blockDim.x + threadIdx.x;
    if (i < total) x[i] = fmaxf(x[i], 0.0f);
}

// ---------------------------------------------------------------------------
// Orchestration
// ---------------------------------------------------------------------------
extern "C" void kernel_launch(void* const* d_in, const int* in_sizes, int n_in,
                              void* d_out, int out_size, void* d_ws, size_t ws_size,
                              hipStream_t stream)
{
    const float* x     = (const float*)d_in[0];   // [N, 512]
    const int*   edges = (const int*)  d_in[1];   // [2, E]
    const float* W1    = (const float*)d_in[2];   // [512, 256]
    const float* b1    = (const float*)d_in[3];   // [256]
    const float* W2    = (const float*)d_in[4];   // [256, 128]
    const float* b2    = (const float*)d_in[5];   // [128]

    const int N  = in_sizes[0] / 512;             // 50000
    const int E  = in_sizes[1] / 2;               // 800000
    const int D0 = 512, D1 = 256, D2 = 128;

    const int* src = edges;                       // edges[0, :]
    const int* dst = edges + E;                   // edges[1, :]

    // Workspace layout: [support1 (N*D1)] [agg1 (N*D1)] ; support2 reuses support1.
    float* support1 = (float*)d_ws;
    float* agg1     = support1 + (size_t)N * D1;
    float* support2 = support1;                   // reuse after scatter1
    float* outp     = (float*)d_out;              // doubles as agg2

    const int TPB = 256;

    // ---- Layer 1 ----
    {   // agg1 = broadcast(b1)
        long long total = (long long)N * D1;
        unsigned blocks = (unsigned)((total + TPB - 1) / TPB);
        gcn_fill_bias<<<blocks, TPB, 0, stream>>>(agg1, b1, total, D1 - 1);
    }
    {   // support1 = x @ W1
        int waves  = ((N + 31) / 32) * (D1 / 64);
        unsigned blocks = (unsigned)((waves + (TPB / 32) - 1) / (TPB / 32));
        gcn_gemm_wmma_f32<false><<<blocks, TPB, 0, stream>>>(x, W1, support1, N, D0, D1);
    }
    {   // agg1[dst] += support1[src]
        long long total = (long long)E * (D1 / 4);
        unsigned blocks = (unsigned)((total + TPB - 1) / TPB);
        gcn_scatter_add<<<blocks, TPB, 0, stream>>>(support1, src, dst, agg1,
                                                    total, /*fShift=*/6, /*fMask=*/63, D1);
    }

    // ---- Layer 2 ----  (x1 = relu(agg1) applied on A-load of GEMM2)
    {   // out = broadcast(b2)
        long long total = (long long)N * D2;
        unsigned blocks = (unsigned)((total + TPB - 1) / TPB);
        gcn_fill_bias<<<blocks, TPB, 0, stream>>>(outp, b2, total, D2 - 1);
    }
    {   // support2 = relu(agg1) @ W2
        int waves  = ((N + 31) / 32) * (D2 / 64);
        unsigned blocks = (unsigned)((waves + (TPB / 32) - 1) / (TPB / 32));
        gcn_gemm_wmma_f32<true><<<blocks, TPB, 0, stream>>>(agg1, W2, support2, N, D1, D2);
    }
    {   // out[dst] += support2[src]
        long long total = (long long)E * (D2 / 4);
        unsigned blocks = (unsigned)((total + TPB - 1) / TPB);
        gcn_scatter_add<<<blocks, TPB, 0, stream>>>(support2, src, dst, outp,
                                                    total, /*fShift=*/5, /*fMask=*/31, D2);
    }
    {   // out = relu(out)
        long long total = (long long)N * D2;
        unsigned blocks = (unsigned)((total + TPB - 1) / TPB);
        gcn_relu<<<blocks, TPB, 0, stream>>>(outp, total);
    }
}